// TripletLoss_58188216926353
// MI455X (gfx1250) — compile-verified
//
#include <hip/hip_runtime.h>
#include <hip/hip_bf16.h>
#include <math.h>

// ---------------------------------------------------------------------------
// TripletLoss on MI455X (gfx1250, wave32).
// Sparse reformulation: the E x E gumbel-top-k matrix is ~all -inf; only
// same-anchor (pos_i, neg_j) pairs are finite.  We enumerate those directly:
//  - V_WMMA_F32_16X16X4_F32 computes the bc_i + bc_j outer-sum per 16x16 tile
//    (gated by a wave-uniform validity ballot, so EXEC is all-ones at WMMA),
//  - TENSOR_LOAD_TO_LDS stages the 32KB bc[] array into LDS once per block
//    (completed with s_wait_tensorcnt), feeding both WMMA operands,
//  - threefry2x32 gumbel noise is generated only at valid entries.
// ---------------------------------------------------------------------------

typedef __attribute__((ext_vector_type(2))) float v2f;
typedef __attribute__((ext_vector_type(8))) float v8f;
typedef __attribute__((ext_vector_type(4))) unsigned int u32x4;
typedef __attribute__((ext_vector_type(4))) int i32x4;
typedef __attribute__((ext_vector_type(8))) int i32x8;

#define EDGES   8192
#define NODES   2048
#define KC      8
#define LKP     16
#define MAXS    400
#define EPSF    1e-9f

#ifndef __has_builtin
#define __has_builtin(x) 0
#endif
#if defined(__AMDGCN__) && __has_builtin(__builtin_amdgcn_tensor_load_to_lds)
#define HAVE_TDM 1
#else
#define HAVE_TDM 0
#endif

// ---------------- bit-exact JAX threefry2x32, key = jax.random.key(42) ------
__device__ __forceinline__ unsigned rotl32(unsigned x, unsigned d) {
    return (x << d) | (x >> (32u - d));
}

// jax.random.uniform(key(42), (E,E)) element at flat index `flat`.
// counts = iota(E*E) split in two halves -> lane0/lane1 of threefry2x32.
__device__ float jax_u01(unsigned flat) {
    const unsigned half = (unsigned)EDGES * (unsigned)EDGES / 2u;  // 2^25
    unsigned x0, x1; int second;
    if (flat < half) { x0 = flat;        x1 = flat + half; second = 0; }
    else             { x0 = flat - half; x1 = flat;        second = 1; }
    unsigned ks[3] = { 0u, 42u, 0u ^ 42u ^ 0x1BD11BDAu };
    x0 += ks[0]; x1 += ks[1];
    const unsigned rotA[4] = { 13u, 15u, 26u, 6u  };
    const unsigned rotB[4] = { 17u, 29u, 16u, 24u };
#pragma unroll
    for (int i = 0; i < 5; ++i) {
        const unsigned* r = (i & 1) ? rotB : rotA;
#pragma unroll
        for (int j = 0; j < 4; ++j) { x0 += x1; x1 = rotl32(x1, r[j]); x1 ^= x0; }
        x0 += ks[(i + 1) % 3];
        x1 += ks[(i + 2) % 3] + (unsigned)(i + 1);
    }
    unsigned bits = second ? x1 : x0;
    return __uint_as_float((bits >> 9) | 0x3f800000u) - 1.0f;
}

// ---------------- tiny setup kernels ----------------------------------------
__global__ void k_init(int* maskp, int* candCnt) {
    if (threadIdx.x == 0) { *maskp = 0; *candCnt = 0; }
}

__global__ void k_clst(const float* __restrict__ clusters, int* __restrict__ clst) {
    int n = blockIdx.x * blockDim.x + threadIdx.x;
    if (n >= NODES) return;
    float best = clusters[n * KC];
    int bi = 0;
#pragma unroll
    for (int k = 1; k < KC; ++k) {
        float v = clusters[n * KC + k];
        if (v > best) { best = v; bi = k; }   // first-max wins, like jnp.argmax
    }
    clst[n] = bi;
}

__global__ void k_mask(const int* __restrict__ clst, const int* __restrict__ kps,
                       int* maskp) {
    if (threadIdx.x < LKP) {
        int c = clst[kps[threadIdx.x]];       // c in [0, 8)
        atomicOr(maskp, 1 << c);
    }
}

__global__ void k_edges(const float* __restrict__ clusters, const int* __restrict__ edges,
                        const int* __restrict__ clst, const int* __restrict__ maskp,
                        int* __restrict__ anchor, int* __restrict__ posF,
                        int* __restrict__ negF, float* __restrict__ bc) {
    int e = blockIdx.x * blockDim.x + threadIdx.x;
    if (e >= EDGES) return;
    int a = edges[2 * e], b = edges[2 * e + 1];
    int c0 = clst[a], c1 = clst[b];
    int mask = *maskp;
    int sel  = ((mask >> c0) & 1) | ((mask >> c1) & 1);
    int sims = (c0 == c1) ? 1 : 0;
    posF[e]  = sel & sims;
    negF[e]  = sel & (sims ^ 1);
    anchor[e] = a;
    float s = 0.0f;
#pragma unroll
    for (int k = 0; k < KC; ++k)
        s += sqrtf(clusters[a * KC + k] * clusters[b * KC + k]);
    bc[e] = s;
}

// ---------------- sparse pair sweep: TDM staging + WMMA outer-sum -----------
__global__ void k_pairs(const int* __restrict__ anchor, const int* __restrict__ posF,
                        const int* __restrict__ negF, const float* __restrict__ bc,
                        float* __restrict__ candVal, int* __restrict__ candIdx,
                        int* candCnt, int cap) {
    const int i0   = blockIdx.x * 16;          // one i-tile per block
    const int wave = threadIdx.x >> 5;         // 8 waves per block
    const int lane = threadIdx.x & 31;
    const int n    = lane & 15;
    const bool hi  = lane >= 16;

    __shared__ float sBcJ[EDGES];              // whole bc[] staged in LDS (32KB)
    __shared__ int   sAnc[16];
    __shared__ int   sPos[16];
    __shared__ int   sAny;

#if HAVE_TDM
    if (wave == 0) {
        // 1-D Tensor-DMA descriptor: count=1, type=image(2), data_size=4B,
        // tensor_dim0 = tile_dim0 = EDGES, strides trivial, no multicast.
        unsigned lds_off = (unsigned)(unsigned long long)(&sBcJ[0]);
        unsigned long long ga = (unsigned long long)bc;
        u32x4 g0;
        g0[0] = 1u;                                        // count=1, user D#
        g0[1] = lds_off;                                   // lds_addr
        g0[2] = (unsigned)(ga & 0xffffffffull);            // global_addr[31:0]
        g0[3] = (unsigned)((ga >> 32) & 0x01ffffffull)     // global_addr[56:32]
              | (2u << 30);                                // type = 2
        i32x8 g1;
        g1[0] = (int)(2u << 16);                           // data_size = 4B
        g1[1] = (int)(((unsigned)EDGES & 0xffffu) << 16);  // tensor_dim0 lo16
        g1[2] = (int)(((unsigned)EDGES >> 16) | (1u << 16)); // dim0 hi16, dim1=1
        g1[3] = (int)((unsigned)EDGES << 16);              // tile_dim0 = EDGES
        g1[4] = 0;                                         // tile_dim1/2 unused
        g1[5] = EDGES;                                     // dim0_stride lo32
        g1[6] = 0;
        g1[7] = 0;
        i32x4 gz = (i32x4)0;                               // groups 2/3 unused
#if defined(__clang_major__) && __clang_major__ >= 23
        i32x8 gz8 = (i32x8)0;
        __builtin_amdgcn_tensor_load_to_lds(g0, g1, gz, gz, gz8, 0);
#else
        __builtin_amdgcn_tensor_load_to_lds(g0, g1, gz, gz, 0);
#endif
        __builtin_amdgcn_s_wait_tensorcnt(0);
    }
#else
    for (int e = threadIdx.x; e < EDGES; e += blockDim.x) sBcJ[e] = bc[e];
#endif
    if (threadIdx.x < 16) {
        sAnc[threadIdx.x] = anchor[i0 + threadIdx.x];
        sPos[threadIdx.x] = posF[i0 + threadIdx.x];
    }
    __syncthreads();
    if (threadIdx.x == 0) {
        int a = 0;
#pragma unroll
        for (int m = 0; m < 16; ++m) a |= sPos[m];
        sAny = a;
    }
    __syncthreads();
    if (!sAny) return;                          // no positive rows in this tile

    for (int jt = wave; jt < EDGES / 16; jt += 8) {
        int j  = jt * 16 + n;
        int aj = anchor[j];
        int ng = negF[j];
        float bcj = sBcJ[j];
        {   // prefetch next j-stride of the anchor stream (global_prefetch_b8)
            int pj = jt * 16 + 8 * 16 + n;
            if (pj < EDGES) __builtin_prefetch(&anchor[pj], 0, 1);
        }

        unsigned vmask = 0u;
#pragma unroll
        for (int r = 0; r < 8; ++r) {
            int M = r + (hi ? 8 : 0);
            if (sPos[M] && ng && (sAnc[M] == aj)) vmask |= 1u << r;
        }
        if (__ballot(vmask != 0u) == 0) continue;  // wave-uniform skip

        // Matrix unit computes D[m,n] = bc_i[m] + bc_j[n] for the whole tile:
        //   A(16x4): col0 = bc_i, col1 = 1;  B(4x16): row0 = 1, row1 = bc_j.
        float bci = sBcJ[i0 + n];       // i-tile bc, lanes 0-15 carry rows
        v2f a, b;
        a.x = hi ? 0.0f : bci;          // A[m,0]  (lanes 16-31 hold K=2,3 -> 0)
        a.y = hi ? 0.0f : 1.0f;         // A[m,1]
        b.x = hi ? 0.0f : 1.0f;         // B[0,n]  (lanes 16-31 hold K=2,3 -> 0)
        b.y = hi ? 0.0f : bcj;          // B[1,n]
        v8f cacc = {};
        v8f d = __builtin_amdgcn_wmma_f32_16x16x4_f32(
            /*neg_a=*/false, a, /*neg_b=*/false, b,
            /*c_mod=*/(short)0, cacc, /*reuse_a=*/false, /*reuse_b=*/false);

#pragma unroll
        for (int r = 0; r < 8; ++r) {
            if (!((vmask >> r) & 1u)) continue;
            float w = 1.0f - 0.5f * d[r];
            if (w <= 0.0f) continue;
            int M = r + (hi ? 8 : 0);
            unsigned flat = (unsigned)(i0 + M) * (unsigned)EDGES + (unsigned)j;
            float u = jax_u01(flat);
            float g = -logf(-logf(u + EPSF) + EPSF);      // gumbel
            float val = logf(fmaxf(w, EPSF)) + g;
            int slot = atomicAdd(candCnt, 1);
            if (slot < cap) { candVal[slot] = val; candIdx[slot] = (int)flat; }
        }
    }
}

// ---------------- top-400 over the sparse candidate list --------------------
__global__ void k_topk(float* __restrict__ candVal, const int* __restrict__ candIdx,
                       const int* candCnt, const float* __restrict__ probas,
                       float* __restrict__ out, int cap) {
    __shared__ float sv[256];
    __shared__ int   sf[256];
    __shared__ int   sp[256];
    __shared__ int   sDone;

    int M = *candCnt;
    if (M > cap) M = cap;
    if (M < 0)   M = 0;

    const float NEG_INF = -__builtin_inff();
    float sum = 0.0f;
    int taken = 0;

    for (int it = 0; it < MAXS; ++it) {
        float bv = NEG_INF; int bf = 0x7fffffff; int bp = -1;
        for (int k = threadIdx.x; k < M; k += 256) {
            float v = candVal[k];
            int   f = candIdx[k];
            if (v > bv || (v == bv && f < bf)) { bv = v; bf = f; bp = k; }
        }
        sv[threadIdx.x] = bv; sf[threadIdx.x] = bf; sp[threadIdx.x] = bp;
        __syncthreads();
        for (int s = 128; s > 0; s >>= 1) {
            if ((int)threadIdx.x < s) {
                float ov = sv[threadIdx.x + s];
                int   of = sf[threadIdx.x + s];
                if (ov > sv[threadIdx.x] ||
                    (ov == sv[threadIdx.x] && of < sf[threadIdx.x])) {
                    sv[threadIdx.x] = ov; sf[threadIdx.x] = of;
                    sp[threadIdx.x] = sp[threadIdx.x + s];
                }
            }
            __syncthreads();
        }
        if (threadIdx.x == 0) {
            if (sp[0] < 0 || !(sv[0] > NEG_INF)) {
                sDone = 1;
            } else {
                sDone = 0;
                candVal[sp[0]] = NEG_INF;                 // remove winner
                int f  = sf[0];
                int i_ = f / EDGES;
                int j_ = f % EDGES;
                float pp = probas[i_], pn = probas[j_];
                sum += logf(pp / (pp + pn));
                ++taken;
            }
        }
        __threadfence_block();
        __syncthreads();
        if (sDone) break;
    }
    if (threadIdx.x == 0)
        out[0] = -(sum / fmaxf((float)taken, 1.0f));
}

// ---------------------------------------------------------------------------
extern "C" void kernel_launch(void* const* d_in, const int* in_sizes, int n_in,
                              void* d_out, int out_size, void* d_ws, size_t ws_size,
                              hipStream_t stream) {
    const float* probas   = (const float*)d_in[0];
    const float* clusters = (const float*)d_in[1];
    const int*   edges    = (const int*)d_in[2];
    const int*   kps      = (const int*)d_in[3];
    float*       out      = (float*)d_out;

    // workspace layout (4-byte units)
    int*   wsI    = (int*)d_ws;
    int*   clst   = wsI;                 // [2048]
    int*   maskp  = wsI + NODES;         // [1]
    int*   cnt    = wsI + NODES + 1;     // [1]
    int*   anchor = wsI + NODES + 4;     // [E]
    int*   posF   = anchor + EDGES;      // [E]
    int*   negF   = posF + EDGES;        // [E]
    float* bc     = (float*)(negF + EDGES);  // [E]

    long long usedB = (long long)((NODES + 4) + 4 * EDGES) * 4;
    long long avail = (long long)ws_size - usedB;
    int cap = 0;
    if (avail > 0) {
        long long c = avail / 8;            // val + idx per candidate
        if (c > (1ll << 22)) c = (1ll << 22);
        cap = (int)c;
    }
    float* candVal = bc + EDGES;
    int*   candIdx = (int*)(candVal + cap);

    k_init <<<1, 64, 0, stream>>>(maskp, cnt);
    k_clst <<<(NODES + 255) / 256, 256, 0, stream>>>(clusters, clst);
    k_mask <<<1, 64, 0, stream>>>(clst, kps, maskp);
    k_edges<<<(EDGES + 255) / 256, 256, 0, stream>>>(clusters, edges, clst, maskp,
                                                     anchor, posF, negF, bc);
    k_pairs<<<EDGES / 16, 256, 0, stream>>>(anchor, posF, negF, bc,
                                            candVal, candIdx, cnt, cap);
    k_topk <<<1, 256, 0, stream>>>(candVal, candIdx, cnt, probas, out, cap);
}